// GCN_29592324669624
// MI455X (gfx1250) — compile-verified
//
#include <hip/hip_runtime.h>
#include <math.h>

typedef __attribute__((ext_vector_type(2))) float v2f;
typedef __attribute__((ext_vector_type(8))) float v8f;

#define N_NODES 50000
#define N_EDGES 800000
#define F_IN    512
#define F_HID   128
#define F_OUT   40

// ---------------------------------------------------------------- utilities
__global__ void gcn_zero_kernel(float* __restrict__ p, int n) {
    int i = blockIdx.x * blockDim.x + threadIdx.x;
    if (i < n) p[i] = 0.0f;
}

// Per-edge degree accumulation for both layers.
// deg1: unit weights, deg2: provided weights. Self-loop (+1) added later.
__global__ void gcn_deg_kernel(const long long* __restrict__ edges,
                               const float* __restrict__ w,
                               float* __restrict__ deg1, float* __restrict__ deg2,
                               int E) {
    int e = blockIdx.x * blockDim.x + threadIdx.x;
    if (e >= E) return;
    int d = (int)edges[E + e];
    atomicAdd(deg1 + d, 1.0f);
    atomicAdd(deg2 + d, w[e]);
}

__global__ void gcn_dinv_kernel(const float* __restrict__ deg1,
                                const float* __restrict__ deg2,
                                float* __restrict__ dinv1, float* __restrict__ dinv2,
                                int n) {
    int i = blockIdx.x * blockDim.x + threadIdx.x;
    if (i >= n) return;
    // self-loop contributes +1 to every degree -> deg > 0 always
    dinv1[i] = rsqrtf(deg1[i] + 1.0f);
    dinv2[i] = rsqrtf(deg2[i] + 1.0f);
}

// ---------------------------------------------------------------- GEMM 1
// H[N,128] = X[N,512] @ W[512,128], fp32 WMMA 16x16x4.
// Block = 256 threads = 8 waves; blockIdx.x = M-tile (16 rows); wave = N-tile.
__global__ void __launch_bounds__(256)
gcn_gemm1_wmma(const float* __restrict__ X, const float* __restrict__ W,
               float* __restrict__ H) {
    const int lane  = threadIdx.x & 31;
    const int wave  = threadIdx.x >> 5;          // 0..7 -> 16-col tile
    const int m0    = blockIdx.x * 16;
    const int halfK = (lane >> 4) * 2;           // 0 (lanes 0-15) or 2 (lanes 16-31)
    const int row   = m0 + (lane & 15);          // A-matrix row for this lane
    const int col   = wave * 16 + (lane & 15);   // B/D column for this lane

    const float* __restrict__ xrow = X + (size_t)row * F_IN;

    v8f acc0 = {};
    v8f acc1 = {};
    #pragma unroll 4
    for (int k = 0; k < F_IN; k += 8) {
        // ---- k-step 0
        v2f a0 = *(const v2f*)(xrow + k + halfK);
        v2f b0;
        b0.x = W[(k + halfK) * F_HID + col];
        b0.y = W[(k + halfK + 1) * F_HID + col];
        acc0 = __builtin_amdgcn_wmma_f32_16x16x4_f32(false, a0, false, b0,
                                                     (short)0, acc0, false, false);
        // ---- k-step 1 (independent accumulator breaks WMMA->WMMA RAW chain)
        v2f a1 = *(const v2f*)(xrow + k + 4 + halfK);
        v2f b1;
        b1.x = W[(k + 4 + halfK) * F_HID + col];
        b1.y = W[(k + 5 + halfK) * F_HID + col];
        acc1 = __builtin_amdgcn_wmma_f32_16x16x4_f32(false, a1, false, b1,
                                                     (short)0, acc1, false, false);
    }
    v8f acc = acc0 + acc1;

    const int rbase = m0 + (lane >> 4) * 8;      // D rows: r / r+8 split by half-wave
    #pragma unroll
    for (int r = 0; r < 8; ++r)
        H[(size_t)(rbase + r) * F_HID + col] = acc[r];
}

// ---------------------------------------------------------------- GEMM 2
// H2[N,40] = X1[N,128] @ W2[128,40]; N padded to 48 (3 col tiles), guarded.
__global__ void __launch_bounds__(96)
gcn_gemm2_wmma(const float* __restrict__ X1, const float* __restrict__ W2,
               float* __restrict__ H2) {
    const int lane  = threadIdx.x & 31;
    const int wave  = threadIdx.x >> 5;          // 0..2
    const int m0    = blockIdx.x * 16;
    const int halfK = (lane >> 4) * 2;
    const int row   = m0 + (lane & 15);
    const int col   = wave * 16 + (lane & 15);
    const bool colOK = (col < F_OUT);

    const float* __restrict__ xrow = X1 + (size_t)row * F_HID;

    v8f acc0 = {};
    v8f acc1 = {};
    #pragma unroll 4
    for (int k = 0; k < F_HID; k += 8) {
        v2f a0 = *(const v2f*)(xrow + k + halfK);
        v2f b0;
        b0.x = colOK ? W2[(k + halfK) * F_OUT + col] : 0.0f;       // value select,
        b0.y = colOK ? W2[(k + halfK + 1) * F_OUT + col] : 0.0f;   // not divergence
        acc0 = __builtin_amdgcn_wmma_f32_16x16x4_f32(false, a0, false, b0,
                                                     (short)0, acc0, false, false);
        v2f a1 = *(const v2f*)(xrow + k + 4 + halfK);
        v2f b1;
        b1.x = colOK ? W2[(k + 4 + halfK) * F_OUT + col] : 0.0f;
        b1.y = colOK ? W2[(k + 5 + halfK) * F_OUT + col] : 0.0f;
        acc1 = __builtin_amdgcn_wmma_f32_16x16x4_f32(false, a1, false, b1,
                                                     (short)0, acc1, false, false);
    }
    v8f acc = acc0 + acc1;

    if (colOK) {
        const int rbase = m0 + (lane >> 4) * 8;
        #pragma unroll
        for (int r = 0; r < 8; ++r)
            H2[(size_t)(rbase + r) * F_OUT + col] = acc[r];
    }
}

// ---------------------------------------------------------------- scatters
// Feature-major: thread t -> edge e = t / F, feature f = t % F. Coalesced
// loads of h[src*F + f] and coalesced atomics on agg[dst*F + f].
__global__ void gcn_scatter1_kernel(const long long* __restrict__ edges,
                                    const float* __restrict__ h,
                                    const float* __restrict__ dinv,
                                    float* __restrict__ agg, int E) {
    int t = blockIdx.x * blockDim.x + threadIdx.x;   // E*128 = 102.4M < 2^31
    if (t >= E * F_HID) return;
    int e = t >> 7;          // /128
    int f = t & (F_HID - 1); // %128
    int s = (int)edges[e];
    int d = (int)edges[E + e];
    float nrm = dinv[s] * dinv[d];
    atomicAdd(agg + (size_t)d * F_HID + f, nrm * h[(size_t)s * F_HID + f]);
}

__global__ void gcn_scatter2_kernel(const long long* __restrict__ edges,
                                    const float* __restrict__ w,
                                    const float* __restrict__ h2,
                                    const float* __restrict__ dinv,
                                    float* __restrict__ agg, int E) {
    int t = blockIdx.x * blockDim.x + threadIdx.x;   // E*40 = 32M
    if (t >= E * F_OUT) return;
    int e = t / F_OUT;
    int f = t - e * F_OUT;
    int s = (int)edges[e];
    int d = (int)edges[E + e];
    float nrm = dinv[s] * w[e] * dinv[d];
    atomicAdd(agg + (size_t)d * F_OUT + f, nrm * h2[(size_t)s * F_OUT + f]);
}

// ---------------------------------------------------------------- pointwise
// x1 = relu(agg1 + dinv1^2 * h1 + b1), written in place over agg1.
__global__ void gcn_node1_kernel(float* __restrict__ agg1,
                                 const float* __restrict__ h1,
                                 const float* __restrict__ dinv1,
                                 const float* __restrict__ b1, int n) {
    int t = blockIdx.x * blockDim.x + threadIdx.x;
    if (t >= n * F_HID) return;
    int i = t >> 7;
    int f = t & (F_HID - 1);
    float di = dinv1[i];
    float v = agg1[t] + di * di * h1[t] + b1[f];
    agg1[t] = fmaxf(v, 0.0f);
}

// z = agg2 + dinv2^2 * h2 + b2 ; out = log_softmax(z) over 40 classes.
__global__ void gcn_final_kernel(const float* __restrict__ agg2,
                                 const float* __restrict__ h2,
                                 const float* __restrict__ dinv2,
                                 const float* __restrict__ b2,
                                 float* __restrict__ out, int n) {
    int i = blockIdx.x * blockDim.x + threadIdx.x;
    if (i >= n) return;
    float d2 = dinv2[i] * dinv2[i];
    float z[F_OUT];
    float m = -3.402823466e38f;
    #pragma unroll
    for (int f = 0; f < F_OUT; ++f) {
        size_t idx = (size_t)i * F_OUT + f;
        z[f] = agg2[idx] + d2 * h2[idx] + b2[f];
        m = fmaxf(m, z[f]);
    }
    float s = 0.0f;
    #pragma unroll
    for (int f = 0; f < F_OUT; ++f) s += expf(z[f] - m);
    float lse = logf(s);
    #pragma unroll
    for (int f = 0; f < F_OUT; ++f)
        out[(size_t)i * F_OUT + f] = z[f] - m - lse;
}

// ---------------------------------------------------------------- launcher
extern "C" void kernel_launch(void* const* d_in, const int* in_sizes, int n_in,
                              void* d_out, int out_size, void* d_ws, size_t ws_size,
                              hipStream_t stream) {
    const float*     X  = (const float*)d_in[0];      // [N,512]
    const long long* E  = (const long long*)d_in[1];  // [2,800000] int64
    const float*     Wt = (const float*)d_in[2];      // [800000]
    const float*     W1 = (const float*)d_in[3];      // [512,128]
    const float*     b1 = (const float*)d_in[4];      // [128]
    const float*     W2 = (const float*)d_in[5];      // [128,40]
    const float*     b2 = (const float*)d_in[6];      // [40]
    float* out = (float*)d_out;

    const int n = N_NODES, e = N_EDGES;

    // workspace layout (floats): 50000*(4 + 128 + 128 + 40 + 40) = 17M = 68 MB
    float* ws    = (float*)d_ws;
    float* deg1  = ws;                      // N
    float* deg2  = deg1 + n;                // N
    float* dinv1 = deg2 + n;                // N
    float* dinv2 = dinv1 + n;               // N
    float* h1    = dinv2 + n;               // N*128
    float* agg1  = h1 + (size_t)n * F_HID;  // N*128 (becomes x1 in place)
    float* h2    = agg1 + (size_t)n * F_HID;// N*40
    float* agg2  = h2 + (size_t)n * F_OUT;  // N*40

    const int B = 256;

    // zero the accumulation buffers (ws is not re-poisoned between replays)
    gcn_zero_kernel<<<(2 * n + B - 1) / B, B, 0, stream>>>(deg1, 2 * n);
    gcn_zero_kernel<<<((n * F_HID) + B - 1) / B, B, 0, stream>>>(agg1, n * F_HID);
    gcn_zero_kernel<<<((n * F_OUT) + B - 1) / B, B, 0, stream>>>(agg2, n * F_OUT);

    // degrees + symmetric normalization terms
    gcn_deg_kernel<<<(e + B - 1) / B, B, 0, stream>>>(E, Wt, deg1, deg2, e);
    gcn_dinv_kernel<<<(n + B - 1) / B, B, 0, stream>>>(deg1, deg2, dinv1, dinv2, n);

    // layer 1: dense WMMA GEMM, edge scatter, self-loop + bias + relu
    gcn_gemm1_wmma<<<n / 16, 256, 0, stream>>>(X, W1, h1);
    gcn_scatter1_kernel<<<(e * F_HID + B - 1) / B, B, 0, stream>>>(E, h1, dinv1, agg1, e);
    gcn_node1_kernel<<<(n * F_HID + B - 1) / B, B, 0, stream>>>(agg1, h1, dinv1, b1, n);

    // layer 2: dense WMMA GEMM (N padded to 48), edge scatter, log_softmax
    gcn_gemm2_wmma<<<n / 16, 96, 0, stream>>>(agg1, W2, h2);
    gcn_scatter2_kernel<<<(e * F_OUT + B - 1) / B, B, 0, stream>>>(E, Wt, h2, dinv2, agg2, e);
    gcn_final_kernel<<<(n + B - 1) / B, B, 0, stream>>>(agg2, h2, dinv2, b2, out, n);
}